// DGCNNEncoder_84138409329083
// MI455X (gfx1250) — compile-verified
//
#include <hip/hip_runtime.h>
#include <hip/hip_bf16.h>

typedef __attribute__((ext_vector_type(2))) float v2f;
typedef __attribute__((ext_vector_type(8))) float v8f;

#define SLOPE 0.2f
#define EPSBN 1e-5f
#define KNN 20
#define NPTS 2048
#define BATCH 16

__device__ __forceinline__ float lrelu(float x){ return x >= 0.f ? x : SLOPE*x; }

__device__ __forceinline__ void atomicMaxF(float* addr, float val){
  if (val >= 0.f) atomicMax((int*)addr, __float_as_int(val));
  else            atomicMin((unsigned int*)addr, __float_as_uint(val));
}
__device__ __forceinline__ void atomicMinF(float* addr, float val){
  if (val >= 0.f) atomicMin((int*)addr, __float_as_int(val));
  else            atomicMax((unsigned int*)addr, __float_as_uint(val));
}

// ---------------------------------------------------------------- utilities
__global__ void fill_kernel(float* p, float v, int n){
  int i = blockIdx.x*blockDim.x + threadIdx.x;
  if (i < n) p[i] = v;
}

__global__ void pad_x_kernel(const float* __restrict__ x, float* __restrict__ xpad){
  int i = blockIdx.x*blockDim.x + threadIdx.x;
  if (i >= BATCH*4*NPTS) return;
  int n = i % NPTS; int c = (i / NPTS) & 3; int b = i / (4*NPTS);
  xpad[i] = (c < 3) ? x[(b*3 + c)*NPTS + n] : 0.f;
}

// Wl = W[:, :C] (zero padded to Cp), Wd = W[:, C:2C] - W[:, :C]
__global__ void split_w_kernel(const float* __restrict__ w, float* __restrict__ Wl,
                               float* __restrict__ Wd, int O, int C, int Cp){
  int i = blockIdx.x*blockDim.x + threadIdx.x;
  if (i >= O*Cp) return;
  int c = i % Cp, o = i / Cp;
  float wl = (c < C) ? w[o*2*C + c]     : 0.f;
  float wr = (c < C) ? w[o*2*C + C + c] : 0.f;
  Wl[i] = wl; Wd[i] = wr - wl;
}

__global__ void colnorm_kernel(const float* __restrict__ xb, int bstride, int Cp,
                               float* __restrict__ sqn){
  int i = blockIdx.x*blockDim.x + threadIdx.x;
  if (i >= BATCH*NPTS) return;
  int n = i % NPTS, b = i / NPTS;
  const float* p = xb + (size_t)b*bstride + n;
  float s = 0.f;
  for (int c = 0; c < Cp; ++c){ float v = p[c*NPTS]; s += v*v; }
  sqn[i] = s;
}

// --------------------------------------------------------------- kNN kernel
// One block = (batch b, 16-row panel). WMMA f32 16x16x4 computes the 16 x 2048
// neg-distance panel into dynamic LDS (128 KB), then per-wave top-20 select.
// All top-k arrays use static indexing only (no scratch spills).
template<int K4>
__global__ __launch_bounds__(256) void knn_kernel(const float* __restrict__ xb, int bstride,
                                                  const float* __restrict__ sqn,
                                                  int* __restrict__ idx){
  extern __shared__ float panel[];                 // 16 * NPTS floats
  const int nt = NPTS/16;
  int b  = blockIdx.x / nt;
  int i0 = (blockIdx.x % nt) * 16;
  int lane = threadIdx.x & 31;
  int wave = threadIdx.x >> 5;                     // 0..7
  int m = lane & 15, half = lane >> 4;
  const float* X  = xb  + (size_t)b*bstride;
  const float* SQ = sqn + b*NPTS;

  // A fragments (rows i0..i0+15, all channels): loop-invariant, fully unrolled
  float a0[K4], a1[K4];
#pragma unroll
  for (int kk = 0; kk < K4; ++kk){
    a0[kk] = X[(4*kk + 2*half    )*NPTS + i0 + m];
    a1[kk] = X[(4*kk + 2*half + 1)*NPTS + i0 + m];
  }
  for (int j0 = wave*16; j0 < NPTS; j0 += 8*16){
    v8f acc = {0.f,0.f,0.f,0.f,0.f,0.f,0.f,0.f};
#pragma unroll
    for (int kk = 0; kk < K4; ++kk){
      v2f A;  A.x  = a0[kk]; A.y = a1[kk];
      v2f Bf; Bf.x = X[(4*kk + 2*half    )*NPTS + j0 + m];
              Bf.y = X[(4*kk + 2*half + 1)*NPTS + j0 + m];
      acc = __builtin_amdgcn_wmma_f32_16x16x4_f32(false, A, false, Bf,
                                                  (short)0, acc, false, false);
    }
    float sqj = SQ[j0 + m];
#pragma unroll
    for (int v = 0; v < 8; ++v){
      int row = v + 8*half;
      panel[row*NPTS + j0 + m] = 2.f*acc[v] - SQ[i0 + row] - sqj;
    }
  }
  __syncthreads();

  // top-20 per row: lane-local sorted lists (static-index bubble insert),
  // then a 32-lane butterfly arg-max merge popping list heads.
  for (int r = wave; r < 16; r += 8){
    float tv[KNN]; int ti[KNN];
#pragma unroll
    for (int t = 0; t < KNN; ++t){ tv[t] = -3.4e38f; ti[t] = 0; }
    const float4* row4 = reinterpret_cast<const float4*>(panel + r*NPTS);
    for (int jj = lane; jj < NPTS/4; jj += 32){
      float4 nd4 = row4[jj];
      int jb = jj*4;
#pragma unroll
      for (int e = 0; e < 4; ++e){
        float cv = (e == 0) ? nd4.x : (e == 1) ? nd4.y : (e == 2) ? nd4.z : nd4.w;
        int   ci = jb + e;
        if (cv > tv[KNN-1]){
#pragma unroll
          for (int t = 0; t < KNN; ++t){
            bool sw = cv > tv[t];
            float pv = tv[t]; int pi = ti[t];
            tv[t] = sw ? cv : pv;  ti[t] = sw ? ci : pi;
            cv    = sw ? pv : cv;  ci    = sw ? pi : ci;
          }
        }
      }
    }
    int* op = idx + ((size_t)b*NPTS + i0 + r)*KNN;
    for (int t = 0; t < KNN; ++t){
      float cv = tv[0]; int ci = ti[0]; int cl = lane;   // propose list head
      for (int off = 16; off >= 1; off >>= 1){
        float ov = __shfl_xor(cv, off);
        int   oi = __shfl_xor(ci, off);
        int   ol = __shfl_xor(cl, off);
        if (ov > cv || (ov == cv && ol < cl)){ cv = ov; ci = oi; cl = ol; }
      }
      bool win = (lane == cl);                           // winner pops its head
#pragma unroll
      for (int q = 0; q < KNN-1; ++q){
        tv[q] = win ? tv[q+1] : tv[q];
        ti[q] = win ? ti[q+1] : ti[q];
      }
      if (win) tv[KNN-1] = -3.4e38f;
      if (lane == 0) op[t] = ci;
    }
  }
}

// ------------------------------------------------- g = Wl.x , t = (Wr-Wl).x
// Output layout (B, N, O) so the gather in edge_stats is coalesced over o.
template<int K4>
__global__ __launch_bounds__(128) void gemm_gt_kernel(const float* __restrict__ xb, int bstride,
                                                      const float* __restrict__ Wl,
                                                      const float* __restrict__ Wd, int O,
                                                      float* __restrict__ G, float* __restrict__ T){
  const int nt = NPTS/16;
  int b  = blockIdx.x / nt;
  int n0 = (blockIdx.x % nt) * 16;
  int lane = threadIdx.x & 31;
  int wave = threadIdx.x >> 5;                     // 0..3
  int m = lane & 15, half = lane >> 4;
  const float* X = xb + (size_t)b*bstride;
  const int Cp = 4*K4;

  float a0[K4], a1[K4];
#pragma unroll
  for (int kk = 0; kk < K4; ++kk){
    a0[kk] = X[(4*kk + 2*half    )*NPTS + n0 + m];
    a1[kk] = X[(4*kk + 2*half + 1)*NPTS + n0 + m];
  }
  for (int o0 = wave*16; o0 < O; o0 += 64){
    v8f accG = {0.f,0.f,0.f,0.f,0.f,0.f,0.f,0.f};
    v8f accT = {0.f,0.f,0.f,0.f,0.f,0.f,0.f,0.f};
#pragma unroll
    for (int kk = 0; kk < K4; ++kk){
      v2f A;  A.x  = a0[kk]; A.y = a1[kk];
      const float* wl = Wl + (size_t)(o0 + m)*Cp + 4*kk + 2*half;
      const float* wd = Wd + (size_t)(o0 + m)*Cp + 4*kk + 2*half;
      v2f Bg; Bg.x = wl[0]; Bg.y = wl[1];
      v2f Bd; Bd.x = wd[0]; Bd.y = wd[1];
      accG = __builtin_amdgcn_wmma_f32_16x16x4_f32(false, A, false, Bg,
                                                   (short)0, accG, false, false);
      accT = __builtin_amdgcn_wmma_f32_16x16x4_f32(false, A, false, Bd,
                                                   (short)0, accT, false, false);
    }
#pragma unroll
    for (int v = 0; v < 8; ++v){
      int n = n0 + v + 8*half;
      size_t off = ((size_t)b*NPTS + n)*O + o0 + m;
      G[off] = accG[v];
      T[off] = accT[v];
    }
  }
}

// -------------------- gather + per-(b,n,o) max/min of y, channel sum/sumsq
__global__ void edge_stats_kernel(const float* __restrict__ G, const float* __restrict__ T,
                                  const int* __restrict__ idx, int O,
                                  float* __restrict__ ymax, float* __restrict__ ymin,
                                  float* __restrict__ chsum, float* __restrict__ chsq){
  __shared__ int sidx[32*KNN];
  int o = threadIdx.x;                             // blockDim.x == O
  const int chunks = NPTS/32;
  int b  = blockIdx.x / chunks;
  int nb = (blockIdx.x % chunks) * 32;
  // cooperative staging of this chunk's 32*20 neighbor indices
  const int* gip = idx + ((size_t)b*NPTS + nb)*KNN;
  for (int i = o; i < 32*KNN; i += blockDim.x) sidx[i] = gip[i];
  __syncthreads();

  const float* Gb = G + (size_t)b*NPTS*O + o;
  float s = 0.f, ss = 0.f;
  for (int ni = 0; ni < 32; ++ni){
    int n = nb + ni;
    size_t base = (size_t)b*NPTS + n;
    float tt = T[base*O + o];
    float mx = -3.4e38f, mn = 3.4e38f;
#pragma unroll
    for (int k = 0; k < KNN; ++k){
      int j = sidx[ni*KNN + k];
      float y = Gb[(size_t)j*O] + tt;
      s += y; ss += y*y;
      mx = fmaxf(mx, y); mn = fminf(mn, y);
    }
    ymax[base*O + o] = mx;
    ymin[base*O + o] = mn;
  }
  atomicAdd(&chsum[o], s);
  atomicAdd(&chsq[o], ss);
}

__global__ void bn_finalize_kernel(const float* __restrict__ chsum, const float* __restrict__ chsq,
                                   const float* __restrict__ gamma, const float* __restrict__ beta,
                                   int O, float invcount,
                                   float* __restrict__ scale, float* __restrict__ shift){
  int o = blockIdx.x*blockDim.x + threadIdx.x;
  if (o >= O) return;
  float mean = chsum[o]*invcount;
  float var  = chsq[o]*invcount - mean*mean;
  float sc   = gamma[o]*rsqrtf(var + EPSBN);
  scale[o] = sc;
  shift[o] = beta[o] - mean*sc;
}

// x_next[b,o,n] = lrelu(sc*(sc>=0 ? ymax : ymin) + shift), into the cat buffer slice
__global__ void bn_apply_max_kernel(const float* __restrict__ ymax, const float* __restrict__ ymin,
                                    const float* __restrict__ scale, const float* __restrict__ shift,
                                    int O, float* __restrict__ outp, int obstride){
  int i = blockIdx.x*blockDim.x + threadIdx.x;
  int total = BATCH*O*NPTS;
  if (i >= total) return;
  int n = i % NPTS; int o = (i/NPTS) % O; int b = i / (O*NPTS);
  float sc = scale[o];
  size_t src = ((size_t)b*NPTS + n)*O + o;
  float v = (sc >= 0.f) ? ymax[src] : ymin[src];
  outp[(size_t)b*obstride + (size_t)o*NPTS + n] = lrelu(sc*v + shift[o]);
}

// ------------------- layer 5: fused GEMM + stats + per-(b,o) max/min, no y5 store
__global__ __launch_bounds__(128) void gemm5_stats_kernel(const float* __restrict__ cat,
                                                          const float* __restrict__ W5,
                                                          float* __restrict__ chsum, float* __restrict__ chsq,
                                                          float* __restrict__ maxb, float* __restrict__ minb){
  const int nt = NPTS/16;
  int b  = blockIdx.x / nt;
  int n0 = (blockIdx.x % nt) * 16;
  int lane = threadIdx.x & 31;
  int wave = threadIdx.x >> 5;
  int m = lane & 15, half = lane >> 4;
  int o0 = blockIdx.y*64 + wave*16;
  const float* X = cat + (size_t)b*320*NPTS;

  v8f acc = {0.f,0.f,0.f,0.f,0.f,0.f,0.f,0.f};
#pragma unroll 8
  for (int kk = 0; kk < 80; ++kk){                 // K = 320 channels
    v2f A;  A.x  = X[(4*kk + 2*half    )*NPTS + n0 + m];
            A.y  = X[(4*kk + 2*half + 1)*NPTS + n0 + m];
    const float* wp = W5 + (size_t)(o0 + m)*320 + 4*kk + 2*half;
    v2f Bf; Bf.x = wp[0]; Bf.y = wp[1];
    acc = __builtin_amdgcn_wmma_f32_16x16x4_f32(false, A, false, Bf,
                                                (short)0, acc, false, false);
  }
  float s = 0.f, ss = 0.f, mx = -3.4e38f, mn = 3.4e38f;
#pragma unroll
  for (int v = 0; v < 8; ++v){
    float y = acc[v];
    s += y; ss += y*y;
    mx = fmaxf(mx, y); mn = fminf(mn, y);
  }
  // lanes l and l+16 own the same output channel o0+(l&15): fold halves
  s  += __shfl_xor(s, 16);
  ss += __shfl_xor(ss, 16);
  mx  = fmaxf(mx, __shfl_xor(mx, 16));
  mn  = fminf(mn, __shfl_xor(mn, 16));
  if (half == 0){
    int o = o0 + m;
    atomicAdd(&chsum[o], s);
    atomicAdd(&chsq[o], ss);
    atomicMaxF(&maxb[b*1024 + o], mx);
    atomicMinF(&minb[b*1024 + o], mn);
  }
}

__global__ void final_kernel(const float* __restrict__ chsum, const float* __restrict__ chsq,
                             const float* __restrict__ g5, const float* __restrict__ b5,
                             const float* __restrict__ maxb, const float* __restrict__ minb,
                             float* __restrict__ out){
  int i = blockIdx.x*blockDim.x + threadIdx.x;
  if (i >= BATCH*1024) return;
  int o = i % 1024;
  const float inv = 1.f/((float)BATCH*NPTS);
  float mean = chsum[o]*inv;
  float var  = chsq[o]*inv - mean*mean;
  float sc   = g5[o]*rsqrtf(var + EPSBN);
  float sh   = b5[o] - mean*sc;
  float v = (sc >= 0.f) ? maxb[i] : minb[i];
  out[i] = lrelu(sc*v + sh);
}

// ------------------------------------------------------------------- driver
extern "C" void kernel_launch(void* const* d_in, const int* in_sizes, int n_in,
                              void* d_out, int out_size, void* d_ws, size_t ws_size,
                              hipStream_t stream){
  (void)in_sizes; (void)n_in; (void)out_size; (void)ws_size;
  const float* x = (const float*)d_in[0];
  const float* W [5] = {(const float*)d_in[1], (const float*)d_in[4], (const float*)d_in[7],
                        (const float*)d_in[10], (const float*)d_in[13]};
  const float* Gm[5] = {(const float*)d_in[2], (const float*)d_in[5], (const float*)d_in[8],
                        (const float*)d_in[11], (const float*)d_in[14]};
  const float* Bt[5] = {(const float*)d_in[3], (const float*)d_in[6], (const float*)d_in[9],
                        (const float*)d_in[12], (const float*)d_in[15]};

  char* ws = (char*)d_ws;
  size_t off = 0;
  auto alloc = [&](size_t bytes) -> char* {
    char* p = ws + off; off += (bytes + 255) & ~(size_t)255; return p;
  };
  float* xpad = (float*)alloc((size_t)BATCH*4*NPTS*4);
  float* cat  = (float*)alloc((size_t)BATCH*320*NPTS*4);
  float* sqn  = (float*)alloc((size_t)BATCH*NPTS*4);
  int*   idx  = (int*)  alloc((size_t)BATCH*NPTS*KNN*4);
  float* G    = (float*)alloc((size_t)BATCH*NPTS*128*4);
  float* T    = (float*)alloc((size_t)BATCH*NPTS*128*4);
  float* ymax = (float*)alloc((size_t)BATCH*NPTS*128*4);
  float* ymin = (float*)alloc((size_t)BATCH*NPTS*128*4);
  float* Wl   = (float*)alloc(128*64*4);
  float* Wd   = (float*)alloc(128*64*4);
  float* chs  = (float*)alloc(1024*4);
  float* chq  = (float*)alloc(1024*4);
  float* scl  = (float*)alloc(1024*4);
  float* shf  = (float*)alloc(1024*4);
  float* maxb = (float*)alloc((size_t)BATCH*1024*4);
  float* minb = (float*)alloc((size_t)BATCH*1024*4);

  pad_x_kernel<<<(BATCH*4*NPTS + 255)/256, 256, 0, stream>>>(x, xpad);

  struct LCfg { const float* xin; int bstride; int C; int Cp; int O; int choff; };
  LCfg L[4] = {
    { xpad,             4*NPTS,   3,  4,  64, 0   },
    { cat,              320*NPTS, 64, 64, 64, 64  },
    { cat + 64*NPTS,    320*NPTS, 64, 64, 64, 128 },
    { cat + 128*NPTS,   320*NPTS, 64, 64, 128, 192 },
  };
  const size_t lds_panel = 16*NPTS*sizeof(float);
  for (int l = 0; l < 4; ++l){
    const LCfg& c = L[l];
    split_w_kernel<<<(c.O*c.Cp + 255)/256, 256, 0, stream>>>(W[l], Wl, Wd, c.O, c.C, c.Cp);
    colnorm_kernel<<<(BATCH*NPTS + 255)/256, 256, 0, stream>>>(c.xin, c.bstride, c.Cp, sqn);
    if (c.Cp == 4){
      knn_kernel<1><<<BATCH*(NPTS/16), 256, lds_panel, stream>>>(c.xin, c.bstride, sqn, idx);
      gemm_gt_kernel<1><<<BATCH*(NPTS/16), 128, 0, stream>>>(c.xin, c.bstride, Wl, Wd,
                                                             c.O, G, T);
    } else {
      knn_kernel<16><<<BATCH*(NPTS/16), 256, lds_panel, stream>>>(c.xin, c.bstride, sqn, idx);
      gemm_gt_kernel<16><<<BATCH*(NPTS/16), 128, 0, stream>>>(c.xin, c.bstride, Wl, Wd,
                                                              c.O, G, T);
    }
    fill_kernel<<<4, 256, 0, stream>>>(chs, 0.f, 1024);
    fill_kernel<<<4, 256, 0, stream>>>(chq, 0.f, 1024);
    edge_stats_kernel<<<BATCH*(NPTS/32), c.O, 0, stream>>>(G, T, idx, c.O,
                                                           ymax, ymin, chs, chq);
    bn_finalize_kernel<<<1, 128, 0, stream>>>(chs, chq, Gm[l], Bt[l], c.O,
                                              1.f/((float)BATCH*NPTS*KNN), scl, shf);
    int tot = BATCH*c.O*NPTS;
    bn_apply_max_kernel<<<(tot + 255)/256, 256, 0, stream>>>(ymax, ymin, scl, shf, c.O,
                                                             cat + (size_t)c.choff*NPTS,
                                                             320*NPTS);
  }

  fill_kernel<<<4, 256, 0, stream>>>(chs, 0.f, 1024);
  fill_kernel<<<4, 256, 0, stream>>>(chq, 0.f, 1024);
  fill_kernel<<<(BATCH*1024 + 255)/256, 256, 0, stream>>>(maxb, -3.4e38f, BATCH*1024);
  fill_kernel<<<(BATCH*1024 + 255)/256, 256, 0, stream>>>(minb,  3.4e38f, BATCH*1024);
  dim3 g5grid(BATCH*(NPTS/16), 1024/64);
  gemm5_stats_kernel<<<g5grid, 128, 0, stream>>>(cat, W[4], chs, chq, maxb, minb);
  final_kernel<<<(BATCH*1024 + 255)/256, 256, 0, stream>>>(chs, chq, Gm[4], Bt[4],
                                                           maxb, minb, (float*)d_out);
}